// KNRM_30245159698677
// MI455X (gfx1250) — compile-verified
//
#include <hip/hip_runtime.h>

typedef __attribute__((ext_vector_type(2))) float v2f;
typedef __attribute__((ext_vector_type(8))) float v8f;

#define D_EMB 300
#define L_Q   32
#define R_D   128
#define KNUM  21
#define SD    302           // padded LDS row stride (floats): 1208 B, 8B-aligned
#define NROWS (L_Q + R_D)   // 160
#define NK4   (D_EMB / 4)   // 75 k-steps of 4

// VALU cross-lane add via DPP16 ROW_XMASK (0x160+m): xor within 16-lane row.
// No LDS round-trip, no s_wait_dscnt (unlike __shfl_xor -> ds_bpermute).
#define DPP_XOR_ADD(x, m)                                                    \
    ((x) + __int_as_float(__builtin_amdgcn_update_dpp(                       \
               0, __float_as_int(x), 0x160 + (m), 0xF, 0xF, true)))

// Cross-row (xor 16) add via v_permlanex16 with identity selects.
#define PERMX16_ADD(x)                                                       \
    ((x) + __int_as_float(__builtin_amdgcn_permlanex16(                      \
               __float_as_int(x), __float_as_int(x),                         \
               0x76543210, 0xfedcba98, true, false)))

__launch_bounds__(256, 1)
__global__ void knrm_fused(const float* __restrict__ emb,
                           const float* __restrict__ w,
                           const float* __restrict__ bias,
                           const int* __restrict__ q1,
                           const int* __restrict__ d1,
                           const int* __restrict__ q2,
                           const int* __restrict__ d2,
                           float* __restrict__ out)
{
    // 160 rows * 302 floats * 4B = 193,280 B  (fits CDNA5's 320KB WGP LDS)
    __shared__ float rows[NROWS * SD];
    __shared__ float S[L_Q * KNUM];     // per-query-token kernel sums over R
    __shared__ float km[KNUM];
    __shared__ float logitP[2];

    const int tid  = threadIdx.x;
    const int lane = tid & 31;
    const int wv   = tid >> 5;      // wave 0..7
    const int hf   = lane >> 4;     // half-wave: selects K-pair (A/B) and M+8 (C/D)
    const int r    = lane & 15;     // row-within-tile / N index
    const int bidx = blockIdx.x;

    // wave -> (m-tile, two n-tiles); tiles in a wave share the A fragment
    const int mt  = wv & 1;
    const int nt0 = (wv >> 1) * 2;
    const int nt1 = nt0 + 1;

    for (int pair = 0; pair < 2; ++pair) {
        const int* q = pair ? q2 : q1;
        const int* d = pair ? d2 : d1;

        // zero the pooling accumulator
        for (int i = tid; i < L_Q * KNUM; i += 256) S[i] = 0.0f;

        // ---- stage + L2-normalize 160 gathered rows (one wave per row) ----
        for (int t = wv; t < NROWS; t += 8) {
            const int tok = (t < L_Q) ? q[bidx * L_Q + t]
                                      : d[bidx * R_D + (t - L_Q)];
            const float* src = emb + (long long)tok * D_EMB;
            float v[10];
            float ss = 0.0f;
#pragma unroll
            for (int i = 0; i < 10; ++i) {
                int c = lane + 32 * i;
                v[i] = (c < D_EMB) ? src[c] : 0.0f;
                ss += v[i] * v[i];
            }
            // wave32 sum: 4x DPP row_xmask (within 16) + 1x permlanex16 (across)
            ss = DPP_XOR_ADD(ss, 1);
            ss = DPP_XOR_ADD(ss, 2);
            ss = DPP_XOR_ADD(ss, 4);
            ss = DPP_XOR_ADD(ss, 8);
            ss = PERMX16_ADD(ss);
            const float sc = 1.0f / fmaxf(sqrtf(ss), 1e-12f);  // x / max(||x||,1e-12)
#pragma unroll
            for (int i = 0; i < 10; ++i) {
                int c = lane + 32 * i;
                if (c < D_EMB) rows[t * SD + c] = v[i] * sc;
            }
        }
        __syncthreads();

        // ---- mm = eq(32xD) * ed(128xD)^T via V_WMMA_F32_16X16X4_F32 ----
        // A frag (16x4 f32): lanes 0-15 hold K=k0,k0+1; lanes 16-31 hold K=k0+2,k0+3
        // B frag (4x16 = ed^T): identical per-lane pattern with doc-row base
        const float* A  = &rows[(mt * 16 + r) * SD + 2 * hf];
        const float* B0 = &rows[(L_Q + nt0 * 16 + r) * SD + 2 * hf];
        const float* B1 = &rows[(L_Q + nt1 * 16 + r) * SD + 2 * hf];
        v8f c0 = {0.0f, 0.0f, 0.0f, 0.0f, 0.0f, 0.0f, 0.0f, 0.0f};
        v8f c1 = c0;
#pragma unroll 5
        for (int ks = 0; ks < NK4; ++ks) {
            v2f a  = *(const v2f*)(A  + 4 * ks);
            v2f b0 = *(const v2f*)(B0 + 4 * ks);
            v2f b1 = *(const v2f*)(B1 + 4 * ks);
            c0 = __builtin_amdgcn_wmma_f32_16x16x4_f32(
                     false, a, false, b0, (short)0, c0, false, false);
            c1 = __builtin_amdgcn_wmma_f32_16x16x4_f32(
                     false, a, false, b1, (short)0, c1, false, false);
        }

        // ---- RBF kernel pooling on accumulator registers ----
        // C/D layout: VGPR j -> M = j + 8*hf, N = r; sum over N (doc dim)
        for (int k = 0; k < KNUM; ++k) {
            const float mu   = (k == KNUM - 1) ? 1.0f : (2.0f * k - 19.0f) * 0.05f;
            const float coef = (k == KNUM - 1) ? -500000.0f : -50.0f;  // -0.5/sigma^2
#pragma unroll
            for (int j = 0; j < 8; ++j) {
                float t0 = c0[j] - mu;
                float t1 = c1[j] - mu;
                float e  = __expf(coef * t0 * t0) + __expf(coef * t1 * t1);
                // 16-lane butterfly, pure VALU (each half reduces independently)
                e = DPP_XOR_ADD(e, 1);
                e = DPP_XOR_ADD(e, 2);
                e = DPP_XOR_ADD(e, 4);
                e = DPP_XOR_ADD(e, 8);
                if (r == 0) {
                    int mrow = mt * 16 + j + 8 * hf;
                    atomicAdd(&S[mrow * KNUM + k], e);   // ds_add_f32 (no wait)
                }
            }
        }
        __syncthreads();

        // ---- log1p over doc-sum, sum over query tokens, linear layer ----
        if (tid < KNUM) {
            float acc = 0.0f;
            for (int l = 0; l < L_Q; ++l) acc += log1pf(S[l * KNUM + tid]);
            km[tid] = acc;
        }
        __syncthreads();
        if (tid == 0) {
            float logit = bias[0];
            for (int k = 0; k < KNUM; ++k) logit += km[k] * w[k];
            logitP[pair] = logit;
        }
        __syncthreads();
    }

    if (tid == 0) {
        float x = logitP[0] - logitP[1];
        out[bidx] = 1.0f / (1.0f + __expf(-x));   // sigmoid(logit1 - logit2)
    }
}

extern "C" void kernel_launch(void* const* d_in, const int* in_sizes, int n_in,
                              void* d_out, int out_size, void* d_ws, size_t ws_size,
                              hipStream_t stream) {
    const float* emb  = (const float*)d_in[0];
    const float* w    = (const float*)d_in[1];
    const float* bias = (const float*)d_in[2];
    const int*   q1   = (const int*)d_in[3];
    const int*   dd1  = (const int*)d_in[4];
    const int*   q2   = (const int*)d_in[5];
    const int*   dd2  = (const int*)d_in[6];
    float*       out  = (float*)d_out;

    const int B = in_sizes[3] / L_Q;   // 1024
    knrm_fused<<<B, 256, 0, stream>>>(emb, w, bias, q1, dd1, q2, dd2, out);
    (void)d_ws; (void)ws_size; (void)n_in; (void)out_size;
}